// L2Loss_82042465288471
// MI455X (gfx1250) — compile-verified
//
#include <hip/hip_runtime.h>

typedef __attribute__((ext_vector_type(2))) float v2f;
typedef __attribute__((ext_vector_type(8))) float v8f;

#define S_   8
#define I_   64
#define NRAD 12

// ---------------------------------------------------------------------------
// Prep: d0 = input_l0 - target_l0 (6144 f32), d1 = input_l1 - target_l1
// (18432 f32), and zero the scalar output. Runs every call (graph-safe).
// ---------------------------------------------------------------------------
__global__ __launch_bounds__(256) void prep_kernel(
    const float* __restrict__ in0, const float* __restrict__ in1,
    const float* __restrict__ tg0, const float* __restrict__ tg1,
    float* __restrict__ d0v, float* __restrict__ d1v, float* __restrict__ out) {
  const int t = blockIdx.x * blockDim.x + threadIdx.x;
  const int n0 = S_ * I_ * 1 * NRAD;  // 6144
  const int n1 = S_ * I_ * 3 * NRAD;  // 18432
  if (t < n0) d0v[t] = in0[t] - tg0[t];
  if (t < n1) d1v[t] = in1[t] - tg1[t];
  if (t == 0) *out = 0.0f;
}

// ---------------------------------------------------------------------------
// Per-block-type body. TYPE 0: ovlp_00 (M1=1,M2=1), TYPE 1: ovlp_01
// (M1=1,M2=3, counted twice), TYPE 2: ovlp_11 (M1=3,M2=3).
//
// One wave owns 16 left-rows (i,m,n). It streams O through the A operand of
// V_WMMA_F32_16X16X4_F32; B is the right-hand d-vector (staged in LDS)
// broadcast across the 16 columns; the C accumulator carries
// y[row] = sum_{j,p,q} O[row;j,p,q] * b[j,p,q] across the whole contraction.
// Finally lanes 0/16 (column 0 of each row-half) dot y with a[row].
// ---------------------------------------------------------------------------
template <int TYPE>
__device__ __forceinline__ void loss_body(
    const float* __restrict__ O, const float* __restrict__ av,
    const float* __restrict__ bvg, float* __restrict__ out,
    float* bsh, int bidLocal) {
  constexpr int   M2          = (TYPE == 0) ? 1 : 3;
  constexpr int   M1          = (TYPE == 2) ? 3 : 1;
  constexpr int   BLK         = M1 * M2 * NRAD * NRAD;   // 144 / 432 / 1296
  constexpr int   TILES_PER_S = (I_ * M1 * NRAD) / 16;   // 48 / 48 / 144
  constexpr int   BSTR        = M2 * NRAD;               // 12 / 36 / 36
  constexpr float SCALE       = (TYPE == 1) ? 2.0f : 1.0f;

  const int lane = threadIdx.x & 31;
  // Force wave id scalar so all loop control stays in SALU (EXEC untouched).
  const int wid  = __builtin_amdgcn_readfirstlane(bidLocal * 8 + (threadIdx.x >> 5));
  const int s    = wid / TILES_PER_S;
  const int tile = wid % TILES_PER_S;

  // Stage right-hand d vector for this structure into LDS (all 8 waves of the
  // block share the same s: TILES_PER_S is a multiple of 8).
  for (int t = threadIdx.x; t < I_ * BSTR; t += 256)
    bsh[t] = bvg[s * (I_ * BSTR) + t];
  __syncthreads();

  // A-layout row for this lane (ISA 32-bit A 16x4: lanes l and l+16 hold the
  // same row M=l&15; lane half selects K sub-chunk {q0,q0+1} vs {q0+2,q0+3}).
  const int rA = tile * 16 + (lane & 15);
  int i, m, n;
  if (TYPE == 2) { i = rA / 36; int t2 = rA % 36; m = t2 / 12; n = t2 % 12; }
  else           { i = rA / 12; m = 0;            n = rA % 12; }
  const int qsel = (lane >> 4) << 1;  // 0 for lanes 0-15, 2 for lanes 16-31

  const float* Orow =
      O + (size_t)(s * I_ + i) * I_ * BLK + m * (M2 * NRAD * NRAD) + n * NRAD + qsel;
  const float* blane = bsh + qsel;

  v8f acc = {};
  for (int j = 0; j < I_; ++j) {
    const float* Oj = Orow + (size_t)j * BLK;
    const float* bj = blane + j * BSTR;
#pragma unroll
    for (int p = 0; p < M2; ++p) {
#pragma unroll
      for (int q0 = 0; q0 < NRAD; q0 += 4) {
        v2f A = *(const v2f*)(Oj + p * (NRAD * NRAD) + q0);  // global, streamed once
        v2f B = *(const v2f*)(bj + p * NRAD + q0);           // LDS broadcast
        // 8 args: (neg_a, A, neg_b, B, c_mod, C, reuse_a, reuse_b)
        acc = __builtin_amdgcn_wmma_f32_16x16x4_f32(
            false, A, false, B, (short)0, acc, false, false);
      }
    }
  }

  // All 16 columns of D are identical (B column-broadcast). Lane l holds
  // D[v + 8*(l>=16)][l&15] in acc[v]; use column-0 lanes of each half.
  if ((lane & 15) == 0) {
    const int half = lane >> 4;
    float partial = 0.0f;
#pragma unroll
    for (int v = 0; v < 8; ++v) {
      const int rc = tile * 16 + v + 8 * half;
      float aval;
      if (TYPE == 2) {
        int ai = rc / 36, t2 = rc % 36;
        aval = av[(s * I_ + ai) * 36 + (t2 / 12) * NRAD + (t2 % 12)];
      } else {
        aval = av[(s * I_ + rc / 12) * NRAD + (rc % 12)];
      }
      partial += acc[v] * aval;
    }
    atomicAdd(out, partial * SCALE);
  }
}

// Single launch keeps all three block types resident concurrently:
// blocks [0,48) -> ovlp_00, [48,96) -> ovlp_01, [96,240) -> ovlp_11.
__global__ __launch_bounds__(256) void loss_wmma_kernel(
    const float* __restrict__ O00, const float* __restrict__ O01,
    const float* __restrict__ O11, const float* __restrict__ d0v,
    const float* __restrict__ d1v, float* __restrict__ out) {
  __shared__ __align__(16) float bsh[I_ * 3 * NRAD];  // 2304 floats (9 KB)
  const int bid = blockIdx.x;
  if (bid < 48) {
    loss_body<0>(O00, d0v, d0v, out, bsh, bid);
  } else if (bid < 96) {
    loss_body<1>(O01, d0v, d1v, out, bsh, bid - 48);
  } else {
    loss_body<2>(O11, d1v, d1v, out, bsh, bid - 96);
  }
}

// ---------------------------------------------------------------------------
// d_in order: 0 input_l0, 1 input_l1, 2 target_l0, 3 target_l1,
//             4 ovlp_00, 5 ovlp_01, 6 ovlp_11.   d_out: 1 float (scalar loss).
// ---------------------------------------------------------------------------
extern "C" void kernel_launch(void* const* d_in, const int* in_sizes, int n_in,
                              void* d_out, int out_size, void* d_ws, size_t ws_size,
                              hipStream_t stream) {
  const float* in0 = (const float*)d_in[0];
  const float* in1 = (const float*)d_in[1];
  const float* tg0 = (const float*)d_in[2];
  const float* tg1 = (const float*)d_in[3];
  const float* O00 = (const float*)d_in[4];
  const float* O01 = (const float*)d_in[5];
  const float* O11 = (const float*)d_in[6];

  float* ws  = (float*)d_ws;
  float* d0v = ws;            // 6144 floats
  float* d1v = ws + 6144;     // 18432 floats
  float* out = (float*)d_out;

  // 18432 elements max -> 72 blocks of 256
  prep_kernel<<<72, 256, 0, stream>>>(in0, in1, tg0, tg1, d0v, d1v, out);

  // 240 blocks x 8 waves = 1920 waves (8*48 + 8*48 + 8*144 row-tiles)
  loss_wmma_kernel<<<240, 256, 0, stream>>>(O00, O01, O11, d0v, d1v, out);
}